// ButterflyConv_v1_15375982920005
// MI455X (gfx1250) — compile-verified
//
#include <hip/hip_runtime.h>

#define C64   64
#define HW    192
#define NEG_SLOPE 0.05f

typedef float v2f __attribute__((ext_vector_type(2)));
typedef float v4f __attribute__((ext_vector_type(4)));

__device__ __forceinline__ float leaky(float v) {
    return v > 0.0f ? v : NEG_SLOPE * v;
}

// ---- CDNA5 async global->LDS copies (ASYNCcnt) ----
__device__ __forceinline__ void async_g2l_b32(const float* gsrc, float* ldst) {
    asm volatile("global_load_async_to_lds_b32 %0, %1, off"
                 :
                 : "v"((unsigned)(unsigned long long)(uintptr_t)ldst),
                   "v"((unsigned long long)(uintptr_t)gsrc)
                 : "memory");
}
__device__ __forceinline__ void async_g2l_b64(const float* gsrc, float* ldst) {
    asm volatile("global_load_async_to_lds_b64 %0, %1, off"
                 :
                 : "v"((unsigned)(unsigned long long)(uintptr_t)ldst),
                   "v"((unsigned long long)(uintptr_t)gsrc)
                 : "memory");
}
__device__ __forceinline__ void wait_async0() {
    asm volatile("s_wait_asynccnt 0" ::: "memory");
}

// ============================================================================
// Primary path: 6-pass butterfly chain. One pass = one depthwise-butterfly
// block. WG = one channel pair {c, c^m} x one 64x64 spatial tile (halo 1).
// Whole tensor (151 MB) stays resident in the 192 MB L2 between passes.
// Inner math on <2 x float> so the backend can form v_pk_fma_f32 (VOP3P).
// ============================================================================
#define TP   64            // output tile side
#define TPH  66            // staged tile side (halo 1)
#define PRS  68            // LDS row stride (floats): keeps 8-col windows 16B aligned
#define NT   (TPH * PRS)   // floats per channel tile (4488)

extern "C" __global__ __launch_bounds__(256)
void bfly_pass(const float* __restrict__ src,
               float* __restrict__ dst,
               const float* __restrict__ resid,   // x for final pass, else null
               const float* __restrict__ Wl,      // 64*9 left-branch weights
               const float* __restrict__ Wr,      // 64*9 right-branch weights
               const float* __restrict__ bl,      // 64 left biases
               const float* __restrict__ br_,     // 64 right biases
               int xm)                             // butterfly xor mask
{
    __shared__ float tile[2 * NT];                // 35,904 B

    const int tid = threadIdx.x;
    const int ts  = blockIdx.x;                   // 0..8 spatial tile
    const int tx  = ts % 3, ty = ts / 3;
    const int pr  = blockIdx.y;                   // 0..31 channel pair
    const int n   = blockIdx.z;                   // batch

    const int low = xm - 1;
    const int c0  = ((pr & ~low) << 1) | (pr & low);   // bit 'xm' clear
    const int c1  = c0 | xm;

    const int gy0 = ty * TP - 1;
    const int gx0 = tx * TP - 1;
    const size_t plane = (size_t)HW * HW;
    const float* s0 = src + ((size_t)n * C64 + c0) * plane;
    const float* s1 = src + ((size_t)n * C64 + c1) * plane;

    // ---- stage both 66x66 channel tiles via async copies, edge-clamped ----
    for (int q = tid; q < 2 * TPH * TPH; q += 256) {
        const int chn = q / (TPH * TPH);
        const int r   = q - chn * (TPH * TPH);
        const int yy  = r / TPH;
        const int xx  = r - yy * TPH;
        int gy = gy0 + yy; gy = gy < 0 ? 0 : (gy > HW - 1 ? HW - 1 : gy);
        int gx = gx0 + xx; gx = gx < 0 ? 0 : (gx > HW - 1 ? HW - 1 : gx);
        const float* gp = (chn ? s1 : s0) + (size_t)gy * HW + gx;
        async_g2l_b32(gp, tile + chn * NT + yy * PRS + xx);
    }
    wait_async0();
    __syncthreads();

    // ---- overwrite clamped halo strips with zeros (SAME zero padding) ----
    if (gy0 < 0)
        for (int q = tid; q < 2 * TPH; q += 256)
            tile[(q / TPH) * NT + (q % TPH)] = 0.0f;                       // row 0
    if (gy0 + TPH > HW)
        for (int q = tid; q < 2 * TPH; q += 256)
            tile[(q / TPH) * NT + (TPH - 1) * PRS + (q % TPH)] = 0.0f;     // row 65
    if (gx0 < 0)
        for (int q = tid; q < 2 * TPH; q += 256)
            tile[(q / TPH) * NT + (q % TPH) * PRS] = 0.0f;                 // col 0
    if (gx0 + TPH > HW)
        for (int q = tid; q < 2 * TPH; q += 256)
            tile[(q / TPH) * NT + (q % TPH) * PRS + (TPH - 1)] = 0.0f;     // col 65
    __syncthreads();

    // ---- compute: thread = (channel-of-pair, 4-row quad, 8-col group) ----
    const int ch = tid >> 7;          // 0 -> c0, 1 -> c1
    const int rq = (tid >> 3) & 15;   // 16 quads of 4 rows
    const int cg = tid & 7;           // 8 groups of 8 cols
    const int oc = ch ? c1 : c0;

    const float* lt = tile + ch * NT;         // left branch: own channel
    const float* rt = tile + (1 - ch) * NT;   // right branch: now[mask][oc] = partner
    const float* wlp = Wl + oc * 9;
    const float* wrp = Wr + oc * 9;
    const float  bLv = bl[oc];
    const float  bRv = br_[oc];

    const int y0t = rq * 4;           // window top tile-row (rows y0t..y0t+5)
    const int x0t = cg * 8;           // window left tile-col (cols x0t..x0t+9)

    float res[4][8];

#pragma unroll
    for (int br2 = 0; br2 < 2; ++br2) {
        const float* bt   = br2 ? rt : lt;
        const float* wp   = br2 ? wrp : wlp;
        const float  bias = br2 ? bRv : bLv;

        v2f wv[9];
#pragma unroll
        for (int k = 0; k < 9; ++k) { const float s = wp[k]; wv[k] = (v2f){s, s}; }

        // 6x10 window as column pairs: E[k]=cols(2k,2k+1), O[k]=cols(2k+1,2k+2)
        v2f E[6][5], O[6][4];
#pragma unroll
        for (int r6 = 0; r6 < 6; ++r6) {
            const float* rowp = bt + (y0t + r6) * PRS + x0t;   // 16B aligned
            const v4f a4 = *(const v4f*)(rowp);
            const v4f b4 = *(const v4f*)(rowp + 4);
            const v2f c2 = *(const v2f*)(rowp + 8);
            E[r6][0] = __builtin_shufflevector(a4, a4, 0, 1);
            E[r6][1] = __builtin_shufflevector(a4, a4, 2, 3);
            E[r6][2] = __builtin_shufflevector(b4, b4, 0, 1);
            E[r6][3] = __builtin_shufflevector(b4, b4, 2, 3);
            E[r6][4] = c2;
            O[r6][0] = __builtin_shufflevector(E[r6][0], E[r6][1], 1, 2);
            O[r6][1] = __builtin_shufflevector(E[r6][1], E[r6][2], 1, 2);
            O[r6][2] = __builtin_shufflevector(E[r6][2], E[r6][3], 1, 2);
            O[r6][3] = __builtin_shufflevector(E[r6][3], E[r6][4], 1, 2);
        }

#pragma unroll
        for (int r = 0; r < 4; ++r) {
#pragma unroll
            for (int k = 0; k < 4; ++k) {          // output column pair
                v2f a = (v2f){bias, bias};
#pragma unroll
                for (int dy = 0; dy < 3; ++dy) {
                    a = __builtin_elementwise_fma(wv[dy * 3 + 0], E[r + dy][k],     a);
                    a = __builtin_elementwise_fma(wv[dy * 3 + 1], O[r + dy][k],     a);
                    a = __builtin_elementwise_fma(wv[dy * 3 + 2], E[r + dy][k + 1], a);
                }
                const float l0 = leaky(a.x);
                const float l1 = leaky(a.y);
                if (br2 == 0) { res[r][2 * k]  = l0; res[r][2 * k + 1]  = l1; }
                else          { res[r][2 * k] += l0; res[r][2 * k + 1] += l1; }
            }
        }
    }

    // ---- store (float4 x2 per row, 32B aligned), optional residual ----
    const int gyb = ty * TP + rq * 4;
    const int gxb = tx * TP + cg * 8;
    float* dbase = dst + ((size_t)n * C64 + oc) * plane;
    const float* rbase = resid ? resid + ((size_t)n * C64 + oc) * plane : (const float*)0;

#pragma unroll
    for (int r = 0; r < 4; ++r) {
        const size_t ro = (size_t)(gyb + r) * HW + gxb;
        if (rbase) {
            const float4 x0 = *(const float4*)(rbase + ro);
            const float4 x1 = *(const float4*)(rbase + ro + 4);
            res[r][0] += x0.x; res[r][1] += x0.y; res[r][2] += x0.z; res[r][3] += x0.w;
            res[r][4] += x1.x; res[r][5] += x1.y; res[r][6] += x1.z; res[r][7] += x1.w;
        }
        float4 v0 = make_float4(res[r][0], res[r][1], res[r][2], res[r][3]);
        float4 v1 = make_float4(res[r][4], res[r][5], res[r][6], res[r][7]);
        *(float4*)(dbase + ro)     = v0;
        *(float4*)(dbase + ro + 4) = v1;
    }
}

// ============================================================================
// Fallback (only if d_ws < 151 MB): single fused kernel, all 6 blocks in LDS.
// ============================================================================
#define TOUT  12
#define TIN   24
#define CS    584
#define NBLK  6

extern "C" __global__ __launch_bounds__(256)
void butterfly_fused(const float* __restrict__ x,
                     const float* __restrict__ W,
                     const float* __restrict__ b,
                     float* __restrict__ out) {
    extern __shared__ float smem[];
    float* buf0 = smem;
    float* buf1 = smem + C64 * CS;

    const int tid = threadIdx.x;
    const int tx  = blockIdx.x & 15;
    const int ty  = blockIdx.x >> 4;
    const int n   = blockIdx.y;
    const int gy0 = ty * TOUT - 6;
    const int gx0 = tx * TOUT - 6;
    const float* xin = x + (size_t)n * C64 * HW * HW;

    const bool interior = (gy0 >= 0) && (gx0 >= 0) &&
                          (gy0 + TIN <= HW) && (gx0 + TIN <= HW);
    if (interior) {
        for (int q = tid; q < C64 * TIN * 12; q += 256) {
            const int c  = q / (TIN * 12);
            const int r  = q - c * (TIN * 12);
            const int yy = r / 12;
            const int xx = (r - yy * 12) * 2;
            const float* gp = xin + ((size_t)c * HW + (gy0 + yy)) * HW + (gx0 + xx);
            async_g2l_b64(gp, buf0 + c * CS + yy * TIN + xx);
        }
        wait_async0();
    } else {
        for (int q = tid; q < C64 * TIN * TIN; q += 256) {
            const int c  = q / (TIN * TIN);
            const int r  = q - c * (TIN * TIN);
            const int yy = r / TIN;
            const int xx = r - yy * TIN;
            const int gy = gy0 + yy, gx = gx0 + xx;
            float v = 0.0f;
            if ((unsigned)gy < (unsigned)HW && (unsigned)gx < (unsigned)HW)
                v = xin[((size_t)c * HW + gy) * HW + gx];
            buf0[c * CS + yy * TIN + xx] = v;
        }
    }
    __syncthreads();

    const int c   = tid >> 2;
    const int sub = tid & 3;
    float* cur = buf0;
    float* nxt = buf1;

#pragma unroll
    for (int i = 0; i < NBLK; ++i) {
        const int S = 22 - 2 * i, off = i + 1, half = S >> 1;
        const int npairs = S * half;
        const int cm = c ^ (32 >> i);
        const float* wlp = W + (2 * i) * C64 * 9 + c * 9;
        const float* wrp = W + (2 * i + 1) * C64 * 9 + c * 9;
        float wL[9], wR[9];
#pragma unroll
        for (int k = 0; k < 9; ++k) { wL[k] = wlp[k]; wR[k] = wrp[k]; }
        const float bL = b[(2 * i) * C64 + c];
        const float bR = b[(2 * i + 1) * C64 + c];
        const float* pl = cur + c * CS;
        const float* pr = cur + cm * CS;
        float*       po = nxt + c * CS;

        for (int p = sub; p < npairs; p += 4) {
            const int y  = p / half;
            const int x2 = (p - y * half) * 2;
            const int yy = y + off, xx = x2 + off;
            float a0 = bL, a1 = bL, r0 = bR, r1 = bR;
#pragma unroll
            for (int dy = 0; dy < 3; ++dy) {
                const float* rl = pl + (yy - 1 + dy) * TIN + (xx - 1);
                const float* rr = pr + (yy - 1 + dy) * TIN + (xx - 1);
                const float l0 = rl[0], l1 = rl[1], l2 = rl[2], l3 = rl[3];
                const float q0 = rr[0], q1 = rr[1], q2 = rr[2], q3 = rr[3];
                const float k0 = wL[dy*3+0], k1 = wL[dy*3+1], k2 = wL[dy*3+2];
                const float m0 = wR[dy*3+0], m1 = wR[dy*3+1], m2 = wR[dy*3+2];
                a0 = fmaf(k0, l0, fmaf(k1, l1, fmaf(k2, l2, a0)));
                a1 = fmaf(k0, l1, fmaf(k1, l2, fmaf(k2, l3, a1)));
                r0 = fmaf(m0, q0, fmaf(m1, q1, fmaf(m2, q2, r0)));
                r1 = fmaf(m0, q1, fmaf(m1, q2, fmaf(m2, q3, r1)));
            }
            float v0 = leaky(a0) + leaky(r0);
            float v1 = leaky(a1) + leaky(r1);
            const int gy = gy0 + yy, gx = gx0 + xx;
            const bool oky = (unsigned)gy < (unsigned)HW;
            v0 = (oky && (unsigned)gx       < (unsigned)HW) ? v0 : 0.0f;
            v1 = (oky && (unsigned)(gx + 1) < (unsigned)HW) ? v1 : 0.0f;
            po[yy * TIN + xx]     = v0;
            po[yy * TIN + xx + 1] = v1;
        }
        __syncthreads();
        float* t = cur; cur = nxt; nxt = t;
    }

    for (int e = tid; e < C64 * TOUT * TOUT; e += 256) {
        const int cc = e / (TOUT * TOUT);
        const int r  = e - cc * (TOUT * TOUT);
        const int yy = r / TOUT, xx = r - yy * TOUT;
        const int gy = ty * TOUT + yy, gx = tx * TOUT + xx;
        const size_t go = (((size_t)n * C64 + cc) * HW + gy) * HW + gx;
        out[go] = cur[cc * CS + (yy + 6) * TIN + (xx + 6)]
                + xin[((size_t)cc * HW + gy) * HW + gx];
    }
}

// ============================================================================
extern "C" void kernel_launch(void* const* d_in, const int* in_sizes, int n_in,
                              void* d_out, int out_size, void* d_ws, size_t ws_size,
                              hipStream_t stream) {
    const float* x = (const float*)d_in[0];
    const float* W = (const float*)d_in[1];
    const float* b = (const float*)d_in[2];
    // d_in[3] (masks) unused: butterfly mask is analytically j ^ (32 >> i)
    float* out = (float*)d_out;
    float* ws  = (float*)d_ws;

    const size_t tensor_bytes = (size_t)16 * C64 * HW * HW * sizeof(float); // 151 MB

    if (ws_size >= tensor_bytes) {
        // 6-pass chain: x -> ws -> out -> ws -> out -> ws -> out(+x)
        dim3 grid(9, 32, 16);
        dim3 block(256);
        const float* srcs[6] = { x, ws, out, ws, out, ws };
        float*       dsts[6] = { ws, out, ws, out, ws, out };
        for (int i = 0; i < 6; ++i) {
            const int xm = 32 >> i;
            const float* resid = (i == 5) ? x : (const float*)0;
            hipLaunchKernelGGL(bfly_pass, grid, block, 0, stream,
                               srcs[i], dsts[i], resid,
                               W + (2 * i) * C64 * 9, W + (2 * i + 1) * C64 * 9,
                               b + (2 * i) * C64,     b + (2 * i + 1) * C64,
                               xm);
        }
    } else {
        dim3 grid(16 * 16, 16);
        dim3 block(256);
        const size_t lds_bytes = (size_t)2 * C64 * CS * sizeof(float);
        hipLaunchKernelGGL(butterfly_fused, grid, block, lds_bytes, stream, x, W, b, out);
    }
}